// SelfAttention_44083544326787
// MI455X (gfx1250) — compile-verified
//
#include <hip/hip_runtime.h>

typedef __attribute__((ext_vector_type(2))) float v2f;
typedef __attribute__((ext_vector_type(8))) float v8f;

#define BD    8        // batch
#define CH    64       // channels
#define NPOS  53248    // 13*64*64 flattened spatial
#define WIN   13       // window size
#define TPOS  208      // positions per block = 16 windows = 13 N-tiles of 16
#define NWIN  16       // windows per block
#define BPB   256      // blocks per batch = NPOS / TPOS
#define VS    212      // sV row stride (floats); 212%64=20 -> conflict-free lane pattern
#define QKC   212      // sQK pair-array padded column count (cols 208..211 zeroed)
#define ATS   17       // attT i-stride (17 coprime 64 -> conflict-free)

__global__ __launch_bounds__(256) void attn13_wmma_kernel(
    const float* __restrict__ x,
    const float* __restrict__ wq, const float* __restrict__ bq,
    const float* __restrict__ wk, const float* __restrict__ bk,
    const float* __restrict__ wv, const float* __restrict__ bv,
    float* __restrict__ out)
{
    // ~135 KB static LDS (CDNA5: up to 320 KB per workgroup)
    // Pair-interleaved layouts: element (k, col) lives at float index
    // ((k>>1)*COLS + col)*2 + (k&1)  ->  a WMMA B-fragment (K = 4s+2h+{0,1})
    // is one aligned float2 at pair index (2s+h).
    __shared__ __align__(16) float sXp [2 * 32 * TPOS];   // x tile, 32 ch-pairs x 208 cols
    __shared__ __align__(16) float sV  [CH * VS];         // v proj, [c][pos], cols 208..211 zeroed
    __shared__ __align__(16) float sQKp[2 * 8 * QKC];     // ch-pairs: 0..3 = Q, 4..7 = K
    __shared__ __align__(16) float sAtp[2 * NWIN * 8 * ATS]; // attT[w][j-pair][i], j>=13 zero

    const int tid  = threadIdx.x;
    const int wave = tid >> 5;
    const int lane = tid & 31;
    const int l16  = lane & 15;
    const int h    = lane >> 4;          // half-wave: K offset 2h, M offset 8h

    const int blk = blockIdx.x;
    const int b0  = blk / BPB;
    const int n0  = (blk % BPB) * TPOS;

    // ---------------- stage 0: global -> LDS x tile (pair layout), zero pads ----------------
    const float* xb = x + (size_t)b0 * CH * NPOS + n0;
    for (int i = tid; i < CH * TPOS; i += 256) {
        int c = i / TPOS, p = i % TPOS;
        sXp[((c >> 1) * TPOS + p) * 2 + (c & 1)] = xb[(size_t)c * NPOS + p];
    }
    {   // zero pads so window-overrun reads are exact zeros
        int r = tid >> 2, cp = tid & 3;                       // sV: 64 rows x 4 pad cols
        sV[r * VS + TPOS + cp] = 0.0f;
        if (tid < 64) {                                       // sQK: 8 pairs x 4 cols x 2 comps
            int pr = tid >> 3, cl = TPOS + ((tid >> 1) & 3), cm = tid & 1;
            sQKp[(pr * QKC + cl) * 2 + cm] = 0.0f;
        }
    }
    __syncthreads();

    // ---------------- stage A: projections via V_WMMA_F32_16X16X4_F32 ----------------
    // jobs 0..12 : [wq;wk] (16x64) x X, one N-tile               -> sQKp
    // jobs 13..38: Wv x X, dual M-tiles (mtp 0..1) x N-tile 0..12 -> sV
    for (int job = wave; job < 39; job += 8) {
        if (job < 13) {
            const int col = job * 16 + l16;
            const float* arow = (l16 < 8) ? (wq + l16 * CH) : (wk + (l16 - 8) * CH);
            v8f acc;
            #pragma unroll
            for (int r = 0; r < 8; ++r) acc[r] = (h == 0) ? bq[r] : bk[r];
            v2f af[16];
            #pragma unroll
            for (int s = 0; s < 16; ++s) {
                int k = 4 * s + 2 * h;
                af[s].x = arow[k]; af[s].y = arow[k + 1];
            }
            #pragma unroll
            for (int s = 0; s < 16; ++s) {
                v2f bf = *(const v2f*)(sXp + 2 * ((2 * s + h) * TPOS + col));
                acc = __builtin_amdgcn_wmma_f32_16x16x4_f32(
                        false, af[s], false, bf, (short)0, acc, false, false);
            }
            #pragma unroll
            for (int r = 0; r < 8; ++r)   // channel c = r + 8h -> pair (r>>1)+4h, comp r&1
                sQKp[(((r >> 1) + 4 * h) * QKC + col) * 2 + (r & 1)] = acc[r];
        } else {
            const int t   = job - 13;
            const int m0  = (t / 13) * 32;          // dual M-tiles: rows m0.. and m0+16..
            const int col = (t % 13) * 16 + l16;
            const float* arow0 = wv + (size_t)(m0 + l16) * CH;
            const float* arow1 = arow0 + 16 * CH;
            v8f acc0, acc1;
            #pragma unroll
            for (int r = 0; r < 8; ++r) {
                acc0[r] = bv[m0 + r + 8 * h];
                acc1[r] = bv[m0 + 16 + r + 8 * h];
            }
            v2f af0[16], af1[16];
            #pragma unroll
            for (int s = 0; s < 16; ++s) {
                int k = 4 * s + 2 * h;
                af0[s].x = arow0[k]; af0[s].y = arow0[k + 1];
                af1[s].x = arow1[k]; af1[s].y = arow1[k + 1];
            }
            #pragma unroll
            for (int s = 0; s < 16; ++s) {
                v2f bf = *(const v2f*)(sXp + 2 * ((2 * s + h) * TPOS + col)); // shared B
                acc0 = __builtin_amdgcn_wmma_f32_16x16x4_f32(
                        false, af0[s], false, bf, (short)0, acc0, false, false);
                acc1 = __builtin_amdgcn_wmma_f32_16x16x4_f32(
                        false, af1[s], false, bf, (short)0, acc1, false, false);
            }
            #pragma unroll
            for (int r = 0; r < 8; ++r) {
                sV[(m0 + r + 8 * h) * VS + col]      = acc0[r];
                sV[(m0 + 16 + r + 8 * h) * VS + col] = acc1[r];
            }
        }
    }
    __syncthreads();

    // ---------------- stage B: scores = Q^T K per window, softmax over j ----------------
    for (int w = wave; w < NWIN; w += 8) {
        const int col = w * WIN + l16;   // <= 210 < QKC
        v8f sc = {0.f,0.f,0.f,0.f,0.f,0.f,0.f,0.f};
        #pragma unroll
        for (int st = 0; st < 2; ++st) {
            // A[i][c] = Q[c][w13+i] : ch-pair 2st+h ; B[c][j] = K[c][w13+j] : pair 4+2st+h
            v2f a  = *(const v2f*)(sQKp + 2 * ((    2 * st + h) * QKC + col));
            v2f bf = *(const v2f*)(sQKp + 2 * ((4 + 2 * st + h) * QKC + col));
            sc = __builtin_amdgcn_wmma_f32_16x16x4_f32(
                    false, a, false, bf, (short)0, sc, false, false);
        }
        // D element (M=i=r+8h, N=j=l16); softmax over j = 16-lane shfl reduction
        #pragma unroll
        for (int r = 0; r < 8; ++r) {
            float v = (l16 < WIN) ? sc[r] : -__builtin_inff();
            float mx = v;
            mx = fmaxf(mx, __shfl_xor(mx, 1));
            mx = fmaxf(mx, __shfl_xor(mx, 2));
            mx = fmaxf(mx, __shfl_xor(mx, 4));
            mx = fmaxf(mx, __shfl_xor(mx, 8));
            float e = __expf(v - mx);               // exact 0 for j >= 13
            float sum = e;
            sum += __shfl_xor(sum, 1);
            sum += __shfl_xor(sum, 2);
            sum += __shfl_xor(sum, 4);
            sum += __shfl_xor(sum, 8);
            // attT[w][j=l16][i=r+8h] in pair layout over j
            sAtp[((w * 8 + (l16 >> 1)) * ATS + (r + 8 * h)) * 2 + (l16 & 1)] = e / sum;
        }
    }
    __syncthreads();

    // ---------------- stage C: out_w = V_w (64x13) * att_w^T, 4 M-tiles share B ----------------
    float* ob = out + (size_t)b0 * CH * NPOS + n0;
    #pragma unroll
    for (int wi = 0; wi < 2; ++wi) {
        const int w    = wave * 2 + wi;
        const int colA = w * WIN;
        v8f o0 = {0.f,0.f,0.f,0.f,0.f,0.f,0.f,0.f};
        v8f o1 = o0, o2 = o0, o3 = o0;
        #pragma unroll
        for (int st = 0; st < 4; ++st) {
            int k = 4 * st + 2 * h;                 // K over j (attT rows 13..15 == 0)
            v2f bf = *(const v2f*)(sAtp + 2 * ((w * 8 + (2 * st + h)) * ATS + l16));
            v2f a0, a1, a2, a3;
            a0.x = sV[(      l16) * VS + colA + k]; a0.y = sV[(      l16) * VS + colA + k + 1];
            a1.x = sV[(16  + l16) * VS + colA + k]; a1.y = sV[(16  + l16) * VS + colA + k + 1];
            a2.x = sV[(32  + l16) * VS + colA + k]; a2.y = sV[(32  + l16) * VS + colA + k + 1];
            a3.x = sV[(48  + l16) * VS + colA + k]; a3.y = sV[(48  + l16) * VS + colA + k + 1];
            o0 = __builtin_amdgcn_wmma_f32_16x16x4_f32(false, a0, false, bf, (short)0, o0, false, false);
            o1 = __builtin_amdgcn_wmma_f32_16x16x4_f32(false, a1, false, bf, (short)0, o1, false, false);
            o2 = __builtin_amdgcn_wmma_f32_16x16x4_f32(false, a2, false, bf, (short)0, o2, false, false);
            o3 = __builtin_amdgcn_wmma_f32_16x16x4_f32(false, a3, false, bf, (short)0, o3, false, false);
        }
        if (l16 < WIN) {
            const int p = colA + l16;
            #pragma unroll
            for (int r = 0; r < 8; ++r) {
                const int ro = r + 8 * h;
                ob[(size_t)(ro     ) * NPOS + p] = o0[r];
                ob[(size_t)(ro + 16) * NPOS + p] = o1[r];
                ob[(size_t)(ro + 32) * NPOS + p] = o2[r];
                ob[(size_t)(ro + 48) * NPOS + p] = o3[r];
            }
        }
    }
}

extern "C" void kernel_launch(void* const* d_in, const int* in_sizes, int n_in,
                              void* d_out, int out_size, void* d_ws, size_t ws_size,
                              hipStream_t stream) {
    const float* x  = (const float*)d_in[0];
    const float* wq = (const float*)d_in[1];
    const float* bq = (const float*)d_in[2];
    const float* wk = (const float*)d_in[3];
    const float* bk = (const float*)d_in[4];
    const float* wv = (const float*)d_in[5];
    const float* bv = (const float*)d_in[6];
    float* out = (float*)d_out;

    dim3 grid(BD * BPB);   // 8 * 256 = 2048 blocks, each = 16 windows
    attn13_wmma_kernel<<<grid, 256, 0, stream>>>(x, wq, bq, wk, bk, wv, bv, out);
}